// EnergyModel_70231305224747
// MI455X (gfx1250) — compile-verified
//
#include <hip/hip_runtime.h>
#include <math.h>

typedef __attribute__((ext_vector_type(2))) float v2f;
typedef __attribute__((ext_vector_type(8))) float v8f;

#define N_ATOMS     8192
#define N_BONDS     8192
#define N_ANGLES    16384
#define N_DIHEDRALS 24576

#define WAVES_PER_BLOCK 8
#define TILES_PER_WAVE  4   // 16-wide col tiles per wave -> wave covers 64 cols

// Per-atom squared norms, recomputed deterministically every launch.
__device__ float g_sq[N_ATOMS];

// ---------------------------------------------------------------------------
// Kernel 0: zero the scalar output accumulator + compute |p_i|^2 (L2-resident)
// ---------------------------------------------------------------------------
__global__ void prep_kernel(const float* __restrict__ pos, float* __restrict__ out) {
    int i = blockIdx.x * blockDim.x + threadIdx.x;
    if (i == 0) out[0] = 0.0f;
    if (i < N_ATOMS) {
        float x = pos[3 * i + 0];
        float y = pos[3 * i + 1];
        float z = pos[3 * i + 2];
        g_sq[i] = x * x + y * y + z * z;
    }
}

// ---------------------------------------------------------------------------
// Shared helper: wave32 shuffle reduce -> LDS -> one atomicAdd per block
// ---------------------------------------------------------------------------
__device__ __forceinline__ void block_accum(float v, float* out) {
    #pragma unroll
    for (int off = 16; off > 0; off >>= 1)
        v += __shfl_down(v, off, 32);          // wave32 reduction
    __shared__ float psum[WAVES_PER_BLOCK];
    const int lane = threadIdx.x & 31;
    const int wave = threadIdx.x >> 5;
    if (lane == 0) psum[wave] = v;
    __syncthreads();
    if (threadIdx.x == 0) {
        float s = 0.0f;
        const int nw = (int)(blockDim.x >> 5);
        for (int w = 0; w < nw; ++w) s += psum[w];
        atomicAdd(out, s);
    }
}

// ---------------------------------------------------------------------------
// Kernel 1: Lennard-Jones over the full 8192x8192 pair matrix.
// Each wave owns one 16-row tile and TILES_PER_WAVE 16-col tiles.
// dot(p_i, p_j) for a 16x16 tile via V_WMMA_F32_16X16X4_F32 (K=4: x,y,z,0).
// sigma/epsilon streamed once with non-temporal loads (512 MiB total -> HBM bound).
// ---------------------------------------------------------------------------
__global__ __launch_bounds__(256) void lj_kernel(const float* __restrict__ pos,
                                                 const float* __restrict__ sigma,
                                                 const float* __restrict__ eps,
                                                 float* __restrict__ out) {
    const int lane = threadIdx.x & 31;
    const int wave = threadIdx.x >> 5;
    const int hi   = lane >> 4;         // 0: lanes 0-15 (K=0,1) | 1: lanes 16-31 (K=2,3)
    const int l15  = lane & 15;

    const int rb  = blockIdx.y * 16;                                        // row tile base
    const int cb0 = (blockIdx.x * WAVES_PER_BLOCK + wave) * (TILES_PER_WAVE * 16);

    // ---- A operand: 16x4 row positions (x,y | z,0) per ISA 32-bit A layout ----
    v2f a;
    {
        const float* p = pos + (size_t)(rb + l15) * 3;
        a.x = hi ? p[2] : p[0];         // K=0 (lo half) / K=2 (hi half)
        a.y = hi ? 0.0f : p[1];         // K=1 (lo half) / K=3 pad (hi half)
    }

    // Row squared norms for this lane's 8 C/D elements (M = v + 8*hi)
    float sqr[8];
    #pragma unroll
    for (int v = 0; v < 8; ++v)
        sqr[v] = g_sq[rb + v + hi * 8];

    float acc = 0.0f;

    for (int t = 0; t < TILES_PER_WAVE; ++t) {
        const int cb = cb0 + t * 16;

        // ---- B operand: 4x16 col positions, mirrored layout ----
        v2f b;
        {
            const float* p = pos + (size_t)(cb + l15) * 3;
            b.x = hi ? p[2] : p[0];
            b.y = hi ? 0.0f : p[1];
        }

        v8f czero = {};
        // D = A x B : 16x16 tile of dot(p_row, p_col), f32 accum
        v8f dotv = __builtin_amdgcn_wmma_f32_16x16x4_f32(
            /*neg_a=*/false, a, /*neg_b=*/false, b,
            /*c_mod=*/(short)0, czero, /*reuse_a=*/false, /*reuse_b=*/false);

        const float  sqc  = g_sq[cb + l15];
        const size_t colg = (size_t)(cb + l15);

        #pragma unroll
        for (int v = 0; v < 8; ++v) {
            const size_t rowg = (size_t)(rb + v + hi * 8);
            const size_t idx  = rowg * (size_t)N_ATOMS + colg;
            // Streamed once, bigger than L2 -> non-temporal loads
            const float sg = __builtin_nontemporal_load(sigma + idx);
            const float ep = __builtin_nontemporal_load(eps + idx);

            float d2   = fmaxf(sqr[v] + sqc - 2.0f * dotv[v], 0.0f);
            float dist = __builtin_amdgcn_sqrtf(d2) + 1e-9f;    // v_sqrt_f32
            float r    = sg * __builtin_amdgcn_rcpf(dist);      // v_rcp_f32
            float r2   = r * r;
            float r6   = r2 * r2 * r2;
            acc = fmaf(4.0f * ep, r6 * r6 - r6, acc);
        }
    }

    block_accum(acc, out);
}

// ---------------------------------------------------------------------------
// Kernel 2: bond energy (8192 bonds)
// ---------------------------------------------------------------------------
__global__ void bond_kernel(const float* __restrict__ pos,
                            const float* __restrict__ kb,
                            const float* __restrict__ r0,
                            const int* __restrict__ bidx,
                            float* __restrict__ out) {
    int i = blockIdx.x * blockDim.x + threadIdx.x;
    float e = 0.0f;
    if (i < N_BONDS) {
        int i0 = bidx[2 * i + 0], i1 = bidx[2 * i + 1];
        float dx = pos[3 * i0 + 0] - pos[3 * i1 + 0];
        float dy = pos[3 * i0 + 1] - pos[3 * i1 + 1];
        float dz = pos[3 * i0 + 2] - pos[3 * i1 + 2];
        float bd = sqrtf(dx * dx + dy * dy + dz * dz) + 1e-9f;
        float d  = bd - r0[i];
        e = 0.5f * kb[i] * d * d;
    }
    block_accum(e, out);
}

// ---------------------------------------------------------------------------
// Kernel 3: angle energy (16384 angles)
// ---------------------------------------------------------------------------
__global__ void angle_kernel(const float* __restrict__ pos,
                             const float* __restrict__ ka,
                             const float* __restrict__ th0,
                             const int* __restrict__ aidx,
                             float* __restrict__ out) {
    int i = blockIdx.x * blockDim.x + threadIdx.x;
    float e = 0.0f;
    if (i < N_ANGLES) {
        int i0 = aidx[3 * i + 0], i1 = aidx[3 * i + 1], i2 = aidx[3 * i + 2];
        float v1x = pos[3 * i1 + 0] - pos[3 * i0 + 0];
        float v1y = pos[3 * i1 + 1] - pos[3 * i0 + 1];
        float v1z = pos[3 * i1 + 2] - pos[3 * i0 + 2];
        float v2x = pos[3 * i1 + 0] - pos[3 * i2 + 0];
        float v2y = pos[3 * i1 + 1] - pos[3 * i2 + 1];
        float v2z = pos[3 * i1 + 2] - pos[3 * i2 + 2];
        float d  = v1x * v2x + v1y * v2y + v1z * v2z;
        float n1 = sqrtf(v1x * v1x + v1y * v1y + v1z * v1z);
        float n2 = sqrtf(v2x * v2x + v2y * v2y + v2z * v2z);
        float c  = d * __builtin_amdgcn_rcpf(n1 * n2);
        c = fminf(1.0f, fmaxf(-1.0f, c));
        float ang = acosf(c);
        float dd  = ang - th0[i];
        e = 0.5f * ka[i] * dd * dd;
    }
    block_accum(e, out);
}

// ---------------------------------------------------------------------------
// Kernel 4: dihedral energy (24576 dihedrals)
// ---------------------------------------------------------------------------
__global__ void dihedral_kernel(const float* __restrict__ pos,
                                const float* __restrict__ kd,
                                const float* __restrict__ ph,
                                const float* __restrict__ nm,
                                const int* __restrict__ didx,
                                float* __restrict__ out) {
    int i = blockIdx.x * blockDim.x + threadIdx.x;
    float e = 0.0f;
    if (i < N_DIHEDRALS) {
        int i0 = didx[4 * i + 0], i1 = didx[4 * i + 1];
        int i2 = didx[4 * i + 2], i3 = didx[4 * i + 3];
        float w1x = pos[3 * i1 + 0] - pos[3 * i0 + 0];
        float w1y = pos[3 * i1 + 1] - pos[3 * i0 + 1];
        float w1z = pos[3 * i1 + 2] - pos[3 * i0 + 2];
        float w2x = pos[3 * i2 + 0] - pos[3 * i1 + 0];
        float w2y = pos[3 * i2 + 1] - pos[3 * i1 + 1];
        float w2z = pos[3 * i2 + 2] - pos[3 * i1 + 2];
        float w3x = pos[3 * i3 + 0] - pos[3 * i2 + 0];
        float w3y = pos[3 * i3 + 1] - pos[3 * i2 + 1];
        float w3z = pos[3 * i3 + 2] - pos[3 * i2 + 2];
        // n1 = w1 x w2 ; n2 = w2 x w3
        float n1x = w1y * w2z - w1z * w2y;
        float n1y = w1z * w2x - w1x * w2z;
        float n1z = w1x * w2y - w1y * w2x;
        float n2x = w2y * w3z - w2z * w3y;
        float n2y = w2z * w3x - w2x * w3z;
        float n2z = w2x * w3y - w2y * w3x;
        // c = n1 x n2
        float cx = n1y * n2z - n1z * n2y;
        float cy = n1z * n2x - n1x * n2z;
        float cz = n1x * n2y - n1y * n2x;
        float lw1 = sqrtf(w1x * w1x + w1y * w1y + w1z * w1z);
        float lw2 = sqrtf(w2x * w2x + w2y * w2y + w2z * w2z);
        float ln1 = sqrtf(n1x * n1x + n1y * n1y + n1z * n1z);
        float ln2 = sqrtf(n2x * n2x + n2y * n2y + n2z * n2z);
        float cos_d = (n1x * n2x + n1y * n2y + n1z * n2z) *
                      __builtin_amdgcn_rcpf(lw1 * lw2);   // (sic: reference uses |w1||w2|)
        float sin_d = (cx * w2x + cy * w2y + cz * w2z) *
                      __builtin_amdgcn_rcpf(lw2 * ln1 * ln2);
        float dih = atan2f(sin_d, cos_d);
        e = 0.5f * kd[i] * (1.0f + cosf(nm[i] * dih - ph[i]));
    }
    block_accum(e, out);
}

// ---------------------------------------------------------------------------
// Launch
// ---------------------------------------------------------------------------
extern "C" void kernel_launch(void* const* d_in, const int* in_sizes, int n_in,
                              void* d_out, int out_size, void* d_ws, size_t ws_size,
                              hipStream_t stream) {
    const float* pos   = (const float*)d_in[0];
    const float* sigma = (const float*)d_in[1];
    const float* eps   = (const float*)d_in[2];
    const float* kb    = (const float*)d_in[3];
    const float* r0    = (const float*)d_in[4];
    const int*   bidx  = (const int*)  d_in[5];
    const float* ka    = (const float*)d_in[6];
    const float* th0   = (const float*)d_in[7];
    const int*   aidx  = (const int*)  d_in[8];
    const float* kd    = (const float*)d_in[9];
    const float* ph    = (const float*)d_in[10];
    const float* nm    = (const float*)d_in[11];   // n_mult is float32 in the reference
    const int*   didx  = (const int*)  d_in[12];
    float* out = (float*)d_out;

    (void)in_sizes; (void)n_in; (void)out_size; (void)d_ws; (void)ws_size;

    prep_kernel<<<(N_ATOMS + 255) / 256, 256, 0, stream>>>(pos, out);

    // 8192 cols / (8 waves * 4 tiles * 16 cols) = 16 blocks in x; 512 row tiles in y
    dim3 grid_lj(N_ATOMS / (WAVES_PER_BLOCK * TILES_PER_WAVE * 16), N_ATOMS / 16);
    lj_kernel<<<grid_lj, 256, 0, stream>>>(pos, sigma, eps, out);

    bond_kernel<<<(N_BONDS + 255) / 256, 256, 0, stream>>>(pos, kb, r0, bidx, out);
    angle_kernel<<<(N_ANGLES + 255) / 256, 256, 0, stream>>>(pos, ka, th0, aidx, out);
    dihedral_kernel<<<(N_DIHEDRALS + 255) / 256, 256, 0, stream>>>(pos, kd, ph, nm, didx, out);
}